// SaptioTemporalNN_A_56307021250961
// MI455X (gfx1250) — compile-verified
//
#include <hip/hip_runtime.h>
#include <cmath>

// Problem constants (match reference)
#define NXC   4096
#define NTC   300
#define NZC   32
#define NRINC 3
#define BC    2048

typedef float v2f __attribute__((ext_vector_type(2)));
typedef float v8f __attribute__((ext_vector_type(8)));

__device__ __forceinline__ float xreduce4(float v) {
    // sum across lanes {q, q+8, q+16, q+24} (wave32)
    v += __shfl_xor(v, 8, 32);
    v += __shfl_xor(v, 16, 32);
    return v;
}

__global__ __launch_bounds__(1024, 1)
void stnn_fused_kernel(const float* __restrict__ relations,  // (NX, 3, NX)
                       const float* __restrict__ factors,    // (NT, NX, NZ)
                       const float* __restrict__ Wdyn,       // (128, 32)
                       const float* __restrict__ bdyn,       // (32,)
                       const int*   __restrict__ t_idx,      // (B,)
                       const int*   __restrict__ x_idx,      // (B,)
                       float*       __restrict__ out)        // (B, 32)
{
    __shared__ __align__(16) float zctx[16][132];  // 16 batch rows x 128 (+4 pad)

    const int tid  = threadIdx.x;
    const int lane = tid & 31;
    const int wave = tid >> 5;    // 0..31
    const int elem = wave >> 1;   // batch element within block (0..15)
    const int half = wave & 1;    // K-split half
    const int b    = blockIdx.x * 16 + elem;

    // zero z_ctx staging tile (LDS content is undefined at launch)
    for (int i = tid; i < 16 * 132; i += 1024) ((float*)zctx)[i] = 0.0f;

    const int t = t_idx[b];
    const int x = x_idx[b];

    const float* facT = factors   + (size_t)t * ((size_t)NXC * NZC);
    const float* relX = relations + (size_t)x * ((size_t)NRINC * NXC);

    __syncthreads();

    // ---------------- phase 1: relation-weighted aggregation ----------------
    // lane layout: sub = lane>>3 -> one of 4 consecutive n rows,
    //              q   = lane&7  -> z quad (z = 4q..4q+3)
    // one b128 load per lane covers a contiguous 512B factors block per wave.
    const int sub = lane >> 3;
    const int q   = lane & 7;
    const int z0  = q * 4;

    float4 a0 = make_float4(0.f, 0.f, 0.f, 0.f);
    float4 a1 = a0, a2 = a0;

    const int nBeg = half * (NXC / 2);
    const int nEnd = nBeg + (NXC / 2);
    #pragma unroll 4
    for (int n0 = nBeg; n0 < nEnd; n0 += 4) {
        const int n = n0 + sub;
        const float4 f = *(const float4*)(facT + (size_t)n * NZC + z0);
        const float r0 = relX[0 * NXC + n];
        const float r1 = relX[1 * NXC + n];
        const float r2 = relX[2 * NXC + n];
        a0.x += r0 * f.x; a0.y += r0 * f.y; a0.z += r0 * f.z; a0.w += r0 * f.w;
        a1.x += r1 * f.x; a1.y += r1 * f.y; a1.z += r1 * f.z; a1.w += r1 * f.w;
        a2.x += r2 * f.x; a2.y += r2 * f.y; a2.z += r2 * f.z; a2.w += r2 * f.w;
    }

    // fold the 4 n-phases
    a0.x = xreduce4(a0.x); a0.y = xreduce4(a0.y); a0.z = xreduce4(a0.z); a0.w = xreduce4(a0.w);
    a1.x = xreduce4(a1.x); a1.y = xreduce4(a1.y); a1.z = xreduce4(a1.z); a1.w = xreduce4(a1.w);
    a2.x = xreduce4(a2.x); a2.y = xreduce4(a2.y); a2.z = xreduce4(a2.z); a2.w = xreduce4(a2.w);

    if (lane < 8) {   // q == lane here
        float* row = &zctx[elem][0];
        atomicAdd(&row[32 + z0 + 0], a0.x);
        atomicAdd(&row[32 + z0 + 1], a0.y);
        atomicAdd(&row[32 + z0 + 2], a0.z);
        atomicAdd(&row[32 + z0 + 3], a0.w);
        atomicAdd(&row[64 + z0 + 0], a1.x);
        atomicAdd(&row[64 + z0 + 1], a1.y);
        atomicAdd(&row[64 + z0 + 2], a1.z);
        atomicAdd(&row[64 + z0 + 3], a1.w);
        atomicAdd(&row[96 + z0 + 0], a2.x);
        atomicAdd(&row[96 + z0 + 1], a2.y);
        atomicAdd(&row[96 + z0 + 2], a2.z);
        atomicAdd(&row[96 + z0 + 3], a2.w);
        if (half == 0) {
            // identity relation: z_ctx[0:32] = factors[t, x, :]
            *(float4*)&row[z0] = *(const float4*)(facT + (size_t)x * NZC + z0);
        }
    }
    __syncthreads();

    // ------------- phase 2: out = tanh(Zctx(16x128) @ Wdyn(128x32) + b) -------------
#if __has_builtin(__builtin_amdgcn_wmma_f32_16x16x4_f32)
    if (wave < 2) {
        const int nt = wave;          // N tile (16 cols each)
        const int m  = lane & 15;     // A-matrix row / B,C,D column
        const int kh = lane >> 4;     // K half within the K=4 chunk
        const int nc = nt * 16 + m;   // output column 0..31
        v8f c = {0.f, 0.f, 0.f, 0.f, 0.f, 0.f, 0.f, 0.f};
        #pragma unroll
        for (int kc = 0; kc < 128; kc += 4) {
            v2f av, bv;
            // A 16x4 f32 layout: lanes 0-15 hold K=kc,kc+1; lanes 16-31 hold K=kc+2,kc+3
            av.x = zctx[m][kc + 2 * kh + 0];
            av.y = zctx[m][kc + 2 * kh + 1];
            // B 4x16 f32: rows striped across lanes, half-wave split on K like C/D
            bv.x = Wdyn[(kc + 2 * kh + 0) * NZC + nc];
            bv.y = Wdyn[(kc + 2 * kh + 1) * NZC + nc];
            c = __builtin_amdgcn_wmma_f32_16x16x4_f32(
                    false, av, false, bv, (short)0, c, false, false);
        }
        const float bias  = bdyn[nc];
        const int   bBase = blockIdx.x * 16;
        #pragma unroll
        for (int v = 0; v < 8; ++v) {
            const int M = v + 8 * kh;   // C/D layout: VGPR v -> rows v and v+8
            out[(size_t)(bBase + M) * NZC + nc] = tanhf(c[v] + bias);
        }
    }
#else
    // fallback: plain FMA for the tiny MLP (keeps kernel correct if builtin absent)
    if (tid < 512) {
        const int M  = tid >> 5;
        const int nc = tid & 31;
        float s = bdyn[nc];
        #pragma unroll 8
        for (int k = 0; k < 128; ++k) s += zctx[M][k] * Wdyn[k * NZC + nc];
        out[(size_t)(blockIdx.x * 16 + M) * NZC + nc] = tanhf(s);
    }
#endif
}

extern "C" void kernel_launch(void* const* d_in, const int* in_sizes, int n_in,
                              void* d_out, int out_size, void* d_ws, size_t ws_size,
                              hipStream_t stream) {
    (void)in_sizes; (void)n_in; (void)d_ws; (void)ws_size; (void)out_size;
    const float* relations = (const float*)d_in[0];
    const float* factors   = (const float*)d_in[1];
    const float* Wdyn      = (const float*)d_in[2];
    const float* bdyn      = (const float*)d_in[3];
    const int*   t_idx     = (const int*)d_in[4];
    const int*   x_idx     = (const int*)d_in[5];
    float* out = (float*)d_out;

    dim3 grid(BC / 16);     // 128 blocks x 16 batch elements
    dim3 block(1024);       // 32 waves: 2 K-split waves per batch element
    hipLaunchKernelGGL(stnn_fused_kernel, grid, block, 0, stream,
                       relations, factors, Wdyn, bdyn, t_idx, x_idx, out);
}